// GraphBlock_309237645785
// MI455X (gfx1250) — compile-verified
//
#include <hip/hip_runtime.h>
#include <hip/hip_bf16.h>
#include <math.h>

// Problem constants (from reference)
#define BCG 128      // B*C graphs
#define NN  2048     // nodes per graph
#define EE  16384    // edges per graph
#define LL  128      // tokens per graph
#define DD  300      // node feature dim
#define HH  256      // hidden after linear1
#define HLL 256      // LSTM hidden
#define G4  1024     // 4*HL

typedef float v2f __attribute__((ext_vector_type(2)));
typedef float v8f __attribute__((ext_vector_type(8)));

__device__ inline v8f wmma_f32x4(v2f a, v2f b, v8f c) {
  // V_WMMA_F32_16X16X4_F32 : D = A(16x4,f32) x B(4x16,f32) + C(16x16,f32)
  return __builtin_amdgcn_wmma_f32_16x16x4_f32(
      /*neg_a=*/false, a, /*neg_b=*/false, b,
      /*c_mod=*/(short)0, c, /*reuse_a=*/false, /*reuse_b=*/false);
}

// ---------------------------------------------------------------------------
// Kernel 1: per-graph token dedup (node -> slot map) + zero accumulators
// ---------------------------------------------------------------------------
__global__ void k_prep(const int* __restrict__ token_ids,
                       int* __restrict__ map, int* __restrict__ tok2slot,
                       float* __restrict__ cnt, float* __restrict__ h_tok) {
  const int bc  = blockIdx.x;
  const int tid = threadIdx.x;               // 128 threads
  int* M = map + bc * NN;
  for (int i = tid; i < NN; i += 128) M[i] = -1;
  __syncthreads();
  if (tid == 0) {
    int ns = 0;
    const int* tok = token_ids + bc * LL;
    int* t2s = tok2slot + bc * LL;
    for (int l = 0; l < LL; ++l) {
      int n = tok[l];
      int s = M[n];
      if (s < 0) { s = ns++; M[n] = s; }
      t2s[l] = s;
    }
  }
  cnt[bc * LL + tid] = 0.0f;
  float* H = h_tok + (size_t)bc * LL * DD;
  for (int i = tid; i < LL * DD; i += 128) H[i] = 0.0f;
}

// ---------------------------------------------------------------------------
// Kernel 2: sparse message accumulation — only edges whose dst is a token node
// One wave per edge; lanes split the D=300 feature accumulation.
// ---------------------------------------------------------------------------
__global__ void k_edges(const float* __restrict__ node_emb,
                        const int* __restrict__ edge_src,
                        const int* __restrict__ edge_dst,
                        const float* __restrict__ edge_val,
                        const int* __restrict__ map,
                        float* __restrict__ cnt, float* __restrict__ h_tok) {
  const int bc   = blockIdx.y;
  const int wave = threadIdx.x >> 5;
  const int lane = threadIdx.x & 31;
  const int e    = blockIdx.x * (blockDim.x >> 5) + wave;   // edge in graph
  const int dst  = edge_dst[bc * EE + e];
  const int slot = map[bc * NN + dst];
  if (slot < 0) return;                                     // wave-uniform
  const int   src = edge_src[bc * EE + e];
  const float val = edge_val[bc * EE + e];
  const float* row = node_emb + ((size_t)bc * NN + src) * DD;
  float* acc = h_tok + ((size_t)bc * LL + slot) * DD;
  if (lane == 0) atomicAdd(&cnt[bc * LL + slot], 1.0f);
  for (int d = lane; d < DD; d += 32) atomicAdd(&acc[d], row[d] * val);
}

// ---------------------------------------------------------------------------
// Kernel 3: linear1 via f32 WMMA. Per graph: [128 x 300] @ [300 x 256] + b1.
// Mean-divide fused as per-row scale on the A fragments.
// ---------------------------------------------------------------------------
__global__ void k_linear1(const float* __restrict__ h_tok,
                          const float* __restrict__ cnt,
                          const float* __restrict__ W1,
                          const float* __restrict__ b1,
                          float* __restrict__ t_tok) {
  const int bc   = blockIdx.x;
  const int lane = threadIdx.x & 31;
  const int wave = threadIdx.x >> 5;          // 8 waves -> 8 M-tiles
  const int m0   = wave * 16;
  const int rA   = lane & 15;                 // A-fragment row
  const int kh   = (lane >> 4) * 2;           // A/B K sub-offset (ISA layout)
  const float* A = h_tok + (size_t)bc * LL * DD;
  float* T       = t_tok + (size_t)bc * LL * HH;
  const float scale = 1.0f / fmaxf(cnt[bc * LL + m0 + rA], 1.0f);
  for (int nt = 0; nt < 16; ++nt) {
    const int nc = nt * 16 + (lane & 15);
    v8f c = {};
    for (int kt = 0; kt < DD / 4; ++kt) {     // 75 k-steps of 4
      const int k0 = kt * 4 + kh;
      v2f a, b;
      a.x = A[(m0 + rA) * DD + k0]     * scale;
      a.y = A[(m0 + rA) * DD + k0 + 1] * scale;
      b.x = W1[(k0)     * HH + nc];
      b.y = W1[(k0 + 1) * HH + nc];
      c = wmma_f32x4(a, b, c);
    }
    const float bias = b1[nc];
#pragma unroll
    for (int j = 0; j < 8; ++j) {
      const int row = m0 + j + 8 * (lane >> 4);
      T[row * HH + nc] = c[j] + bias;
    }
  }
}

// ---------------------------------------------------------------------------
// Kernel 4: scatter token slots into LSTM sequence layout seq[L][BC][H]
// ---------------------------------------------------------------------------
__global__ void k_gather_seq(const float* __restrict__ t_tok,
                             const int* __restrict__ tok2slot,
                             float* __restrict__ seq) {
  const int blk  = blockIdx.x;          // bc*L + l
  const int bc   = blk >> 7;
  const int l    = blk & (LL - 1);
  const int slot = tok2slot[bc * LL + l];
  const float* s = t_tok + ((size_t)bc * LL + slot) * HH;
  float* d       = seq + ((size_t)l * BCG + bc) * HH;
  d[threadIdx.x] = s[threadIdx.x];
}

__global__ void k_zero(float* __restrict__ p, int n) {
  const int i = blockIdx.x * blockDim.x + threadIdx.x;
  if (i < n) p[i] = 0.0f;
}

// ---------------------------------------------------------------------------
// Kernel 5: LSTM gates, one step. Block = 16 rows; LDS-staged [16 x 512] x|h
// tile, 8 waves x 8 N-tiles cover all 1024 gate columns via f32 WMMA.
// ---------------------------------------------------------------------------
__global__ void k_lstm_gates(const float* __restrict__ seq_t,   // [BC][H]
                             const float* __restrict__ h_state, // [BC][HL]
                             const float* __restrict__ W_ih,    // [H][4HL]
                             const float* __restrict__ W_hh,    // [HL][4HL]
                             const float* __restrict__ b_ih,
                             const float* __restrict__ b_hh,
                             float* __restrict__ gates) {       // [BC][4HL]
  __shared__ float xh[16 * 512];                                // 32 KB LDS
  const int m0 = blockIdx.x * 16;
  for (int i = threadIdx.x; i < 16 * 512; i += blockDim.x) {
    const int r = i >> 9, col = i & 511;
    xh[i] = (col < HH) ? seq_t[(m0 + r) * HH + col]
                       : h_state[(m0 + r) * HLL + (col - HH)];
  }
  __syncthreads();
  const int lane = threadIdx.x & 31;
  const int wave = threadIdx.x >> 5;
  const int rA   = lane & 15;
  const int kh   = (lane >> 4) * 2;
  for (int nt = 0; nt < 8; ++nt) {
    const int nc = (wave * 8 + nt) * 16 + (lane & 15);
    v8f c = {};
    for (int kt = 0; kt < HH / 4; ++kt) {        // x_t @ W_ih
      const int k0 = kt * 4 + kh;
      v2f a, b;
      a.x = xh[rA * 512 + k0];
      a.y = xh[rA * 512 + k0 + 1];
      b.x = W_ih[(k0)     * G4 + nc];
      b.y = W_ih[(k0 + 1) * G4 + nc];
      c = wmma_f32x4(a, b, c);
    }
    for (int kt = 0; kt < HLL / 4; ++kt) {       // h @ W_hh
      const int k0 = kt * 4 + kh;
      v2f a, b;
      a.x = xh[rA * 512 + HH + k0];
      a.y = xh[rA * 512 + HH + k0 + 1];
      b.x = W_hh[(k0)     * G4 + nc];
      b.y = W_hh[(k0 + 1) * G4 + nc];
      c = wmma_f32x4(a, b, c);
    }
    const float bias = b_ih[nc] + b_hh[nc];
#pragma unroll
    for (int j = 0; j < 8; ++j) {
      const int row = m0 + j + 8 * (lane >> 4);
      gates[row * G4 + nc] = c[j] + bias;
    }
  }
}

// Kernel 6: LSTM cell update (PyTorch gate order i, f, g, o)
__global__ void k_lstm_update(const float* __restrict__ gates,
                              float* __restrict__ h_state,
                              float* __restrict__ c_state) {
  const int idx = blockIdx.x * blockDim.x + threadIdx.x;
  if (idx >= BCG * HLL) return;
  const int bc = idx >> 8, hl = idx & (HLL - 1);
  const float* g = gates + (size_t)bc * G4;
  const float gi = g[hl], gf = g[HLL + hl], gg = g[2 * HLL + hl], go = g[3 * HLL + hl];
  const float si = 1.0f / (1.0f + __expf(-gi));
  const float sf = 1.0f / (1.0f + __expf(-gf));
  const float so = 1.0f / (1.0f + __expf(-go));
  const float cn = sf * c_state[idx] + si * tanhf(gg);
  c_state[idx] = cn;
  h_state[idx] = so * tanhf(cn);
}

// ---------------------------------------------------------------------------
// Kernel 7: head — W2 projection, softmax over C=4, BCE-style loss reduction
// ---------------------------------------------------------------------------
__global__ void k_head(const float* __restrict__ h_state,
                       const float* __restrict__ W2, const float* __restrict__ b2,
                       const float* __restrict__ labels, float* __restrict__ out) {
  __shared__ float sc[BCG];
  __shared__ float ex[BCG];
  __shared__ float red[BCG];
  const int bc = threadIdx.x;                  // 0..127
  const float* h = h_state + (size_t)bc * HLL;
  float acc = b2[0];
  for (int k = 0; k < HLL; ++k) acc += h[k] * W2[k];
  sc[bc] = acc;
  __syncthreads();
  const int b4 = (bc >> 2) * 4;
  const float m = fmaxf(fmaxf(sc[b4], sc[b4 + 1]), fmaxf(sc[b4 + 2], sc[b4 + 3]));
  const float e = __expf(sc[bc] - m);
  ex[bc] = e;
  __syncthreads();
  const float s = ex[b4] + ex[b4 + 1] + ex[b4 + 2] + ex[b4 + 3];
  const float x = e / s;
  out[1 + bc] = x;                             // soft, flat [B,C]
  const float y = labels[bc];
  red[bc] = fmaxf(x, 0.0f) - x * y + log1pf(__expf(-fabsf(x)));
  __syncthreads();
  for (int st = 64; st > 0; st >>= 1) {
    if (bc < st) red[bc] += red[bc + st];
    __syncthreads();
  }
  if (bc == 0) out[0] = red[0] / (float)BCG;   // error (mean)
}

// ---------------------------------------------------------------------------
extern "C" void kernel_launch(void* const* d_in, const int* in_sizes, int n_in,
                              void* d_out, int out_size, void* d_ws, size_t ws_size,
                              hipStream_t stream) {
  const float* node_emb  = (const float*)d_in[0];
  const int*   edge_src  = (const int*)  d_in[1];
  const int*   edge_dst  = (const int*)  d_in[2];
  const float* edge_val  = (const float*)d_in[3];
  const int*   token_ids = (const int*)  d_in[4];
  const float* labels    = (const float*)d_in[5];
  const float* W1        = (const float*)d_in[6];
  const float* b1        = (const float*)d_in[7];
  const float* W_ih      = (const float*)d_in[8];
  const float* W_hh      = (const float*)d_in[9];
  const float* b_ih      = (const float*)d_in[10];
  const float* b_hh      = (const float*)d_in[11];
  const float* W2        = (const float*)d_in[12];
  const float* b2        = (const float*)d_in[13];
  float* out = (float*)d_out;

  // Workspace carve-up
  char* ws = (char*)d_ws;
  size_t off = 0;
  auto carve = [&](size_t bytes) { void* p = ws + off; off += (bytes + 255) & ~(size_t)255; return p; };
  int*   map      = (int*)  carve((size_t)BCG * NN * sizeof(int));        // 1 MB
  int*   tok2slot = (int*)  carve((size_t)BCG * LL * sizeof(int));
  float* cnt      = (float*)carve((size_t)BCG * LL * sizeof(float));
  float* h_tok    = (float*)carve((size_t)BCG * LL * DD * sizeof(float)); // 19.7 MB
  float* t_tok    = (float*)carve((size_t)BCG * LL * HH * sizeof(float)); // 16.8 MB
  float* seq      = (float*)carve((size_t)LL * BCG * HH * sizeof(float)); // 16.8 MB
  float* h_state  = (float*)carve((size_t)BCG * HLL * sizeof(float));
  float* c_state  = (float*)carve((size_t)BCG * HLL * sizeof(float));
  float* gates    = (float*)carve((size_t)BCG * G4 * sizeof(float));
  (void)ws_size; (void)in_sizes; (void)n_in; (void)out_size;

  // 1) dedup tokens + zero accumulators
  k_prep<<<BCG, 128, 0, stream>>>(token_ids, map, tok2slot, cnt, h_tok);
  // 2) token-sparse message accumulation (one wave per edge)
  k_edges<<<dim3(EE / 8, BCG), 256, 0, stream>>>(node_emb, edge_src, edge_dst,
                                                 edge_val, map, cnt, h_tok);
  // 3) linear1 (WMMA f32), mean fused
  k_linear1<<<BCG, 256, 0, stream>>>(h_tok, cnt, W1, b1, t_tok);
  // 4) scatter into [L][BC][H]
  k_gather_seq<<<BCG * LL, HH, 0, stream>>>(t_tok, tok2slot, seq);
  // 5) zero LSTM state (h_state & c_state are contiguous)
  k_zero<<<(2 * BCG * HLL + 255) / 256, 256, 0, stream>>>(h_state, 2 * BCG * HLL);
  // 6) LSTM recurrence: 128 dependent step pairs on the stream
  for (int t = 0; t < LL; ++t) {
    const float* seq_t = seq + (size_t)t * BCG * HH;
    k_lstm_gates<<<BCG / 16, 256, 0, stream>>>(seq_t, h_state, W_ih, W_hh,
                                               b_ih, b_hh, gates);
    k_lstm_update<<<(BCG * HLL + 255) / 256, 256, 0, stream>>>(gates, h_state, c_state);
  }
  // 7) head: projection + softmax + loss
  k_head<<<1, BCG, 0, stream>>>(h_state, W2, b2, labels, out);
}